// GINE_Regression_51702816309460
// MI455X (gfx1250) — compile-verified
//
#include <hip/hip_runtime.h>

// ---------------------------------------------------------------------------
// GINE regression pipeline for MI455X (gfx1250), fp32 WMMA throughout.
//   N=50000 nodes, E=640000 edges, G=256 graphs, H=128, L=3.
// Round 3: wave-uniform full-tile fast paths for all store/scatter epilogues
// (no per-lane EXEC churn), hoisted src/dst index loads. Inner loops remain:
// 1 coalesced b64 A-load + 1 coalesced b64 packed-B load + 1 v_wmma.
// ---------------------------------------------------------------------------

typedef __attribute__((ext_vector_type(2))) float v2f;
typedef __attribute__((ext_vector_type(8))) float v8f;

#define HDIM 128

// D = A(16x4 f32) * B(4x16 f32) + C(16x16 f32); one wave32.
__device__ __forceinline__ v8f wmma_f32_k4(v2f a, v2f b, v8f c) {
  return __builtin_amdgcn_wmma_f32_16x16x4_f32(false, a, false, b,
                                               (short)0, c, false, false);
}

// B-fragment for (nt, ks) from a packed weight matrix: lane-contiguous pairs.
__device__ __forceinline__ v2f bfrag(const float* __restrict__ P, int KS,
                                     int nt, int ks, int lane) {
  return *(const v2f*)(P + ((((nt * KS + ks) << 5) + lane) << 1));
}

// ---------------------------------------------------------------------------
// Repack W[K x 128] (row-major) into WMMA B-fragment order:
//   P[((nt*KS+ks)*32+lane)*2 + j] = W[(ks*4+(lane>>4)*2+j)*128 + nt*16+(lane&15)]
// ---------------------------------------------------------------------------
__global__ void pack_b_kernel(const float* __restrict__ W,
                              float* __restrict__ P, int K) {
  const int KS = K >> 2;
  const int total = 8 * KS * 32;
  for (int i = blockIdx.x * blockDim.x + threadIdx.x; i < total;
       i += gridDim.x * blockDim.x) {
    const int lane = i & 31;
    const int ks = (i >> 5) % KS;
    const int nt = (i >> 5) / KS;
    const int k = ks * 4 + ((lane >> 4) << 1);
    const int nn = nt * 16 + (lane & 15);
    P[i * 2 + 0] = W[k * HDIM + nn];
    P[i * 2 + 1] = W[(k + 1) * HDIM + nn];
  }
}

// ---------------------------------------------------------------------------
// h[N,128] = x[N,32] @ W[32,128] + b.  One wave per 16-row tile.
// ---------------------------------------------------------------------------
__global__ void node_encode_kernel(const float* __restrict__ x,
                                   const float* __restrict__ pkW,
                                   const float* __restrict__ b,
                                   float* __restrict__ h, int n) {
  const int lane = threadIdx.x & 31;
  const int wid  = threadIdx.x >> 5;
  const int tile = blockIdx.x * (blockDim.x >> 5) + wid;
  const int r0   = tile * 16;
  if (r0 >= n) return;
  const int m    = lane & 15;
  const int half = lane >> 4;
  const bool full = (r0 + 16 <= n);           // wave-uniform
  const int row_a = full ? (r0 + m) : min(r0 + m, n - 1);
#pragma unroll
  for (int nt = 0; nt < 8; ++nt) {
    v8f acc = {};
    const int nn = nt * 16 + m;
#pragma unroll
    for (int ks = 0; ks < 8; ++ks) {        // K = 32 -> 8 steps of 4
      const int k = ks * 4 + half * 2;
      v2f a = *(const v2f*)&x[row_a * 32 + k];
      acc = wmma_f32_k4(a, bfrag(pkW, 8, nt, ks, lane), acc);
    }
    const float bias = b[nn];
    if (full) {
#pragma unroll
      for (int v = 0; v < 8; ++v)
        h[(r0 + v + half * 8) * HDIM + nn] = acc[v] + bias;
    } else {
#pragma unroll
      for (int v = 0; v < 8; ++v) {
        const int row = r0 + v + half * 8;
        if (row < n) h[row * HDIM + nn] = acc[v] + bias;
      }
    }
  }
}

// ---------------------------------------------------------------------------
// Fused per-layer message kernel. For a 16-edge tile:
//   t1 = relu(edge_attr_tile @ ee_w1 + b1)        (WMMA, K=16)
//   e  = t1 @ ee_w2 + b2                          (WMMA, K=128, A via LDS)
//   msg = relu(h[src] + e);  aggr[dst] += msg     (atomic scatter-add)
// 4 waves per block, each wave owns one edge tile + an 8KB LDS slice.
// ---------------------------------------------------------------------------
__global__ void msg_aggr_kernel(const float* __restrict__ ea,
                                const float* __restrict__ pkw1,
                                const float* __restrict__ b1,
                                const float* __restrict__ pkw2,
                                const float* __restrict__ b2,
                                const float* __restrict__ h,
                                const int* __restrict__ src,
                                const int* __restrict__ dst,
                                float* __restrict__ aggr,
                                int nedges, int ntiles) {
  __shared__ float T_all[4][16 * HDIM];
  const int lane = threadIdx.x & 31;
  const int wid  = threadIdx.x >> 5;
  const int et   = blockIdx.x * 4 + wid;
  const bool active = (et < ntiles);
  const int e0   = et * 16;
  const int m    = lane & 15;
  const int half = lane >> 4;
  float* T = T_all[wid];

  // Warm the packed 64KB second-layer weights (global_prefetch_b8).
  __builtin_prefetch(&pkw2[(threadIdx.x & 127) * HDIM], 0, 3);

  if (active) {
    const int er = min(e0 + m, nedges - 1);  // clamped A row
    // ---- stage 1: t1 = relu(ea @ w1 + b1), K=16 ----
#pragma unroll
    for (int nt = 0; nt < 8; ++nt) {
      v8f acc = {};
      const int nn = nt * 16 + m;
#pragma unroll
      for (int ks = 0; ks < 4; ++ks) {
        const int k = ks * 4 + half * 2;
        v2f a = *(const v2f*)&ea[er * 16 + k];
        acc = wmma_f32_k4(a, bfrag(pkw1, 4, nt, ks, lane), acc);
      }
      const float bias = b1[nn];
#pragma unroll
      for (int v = 0; v < 8; ++v) {
        float t = acc[v] + bias;
        T[(v + half * 8) * HDIM + nn] = t > 0.f ? t : 0.f;
      }
    }
  }
  __syncthreads();
  if (active) {
    const bool fullE = (e0 + 16 <= nedges);  // wave-uniform
    // hoist per-tile edge endpoints (row base offsets) out of the nt loop
    int sA[8], dA[8];
#pragma unroll
    for (int v = 0; v < 8; ++v) {
      const int e = min(e0 + v + half * 8, nedges - 1);
      sA[v] = src[e] * HDIM;
      dA[v] = dst[e] * HDIM;
    }
    // ---- stage 2: e = t1 @ w2 + b2, K=128; fused gather/relu/scatter ----
#pragma unroll
    for (int nt = 0; nt < 8; ++nt) {
      v8f acc = {};
      const int nn = nt * 16 + m;
      for (int ks = 0; ks < 32; ++ks) {
        const int k = ks * 4 + half * 2;
        v2f a = *(const v2f*)&T[m * HDIM + k];   // ds_load_b64
        acc = wmma_f32_k4(a, bfrag(pkw2, 32, nt, ks, lane), acc);
      }
      const float bias = b2[nn];
      if (fullE) {
#pragma unroll
        for (int v = 0; v < 8; ++v) {
          float msg = h[sA[v] + nn] + acc[v] + bias;
          msg = msg > 0.f ? msg : 0.f;
          atomicAdd(&aggr[dA[v] + nn], msg);     // global_atomic_add_f32 (L2)
        }
      } else {
#pragma unroll
        for (int v = 0; v < 8; ++v) {
          if (e0 + v + half * 8 < nedges) {
            float msg = h[sA[v] + nn] + acc[v] + bias;
            msg = msg > 0.f ? msg : 0.f;
            atomicAdd(&aggr[dA[v] + nn], msg);
          }
        }
      }
    }
  }
}

// ---------------------------------------------------------------------------
// Per-layer node MLP:  z = relu((h+aggr) @ w1 + b1) @ w2 + b2, plus
// per-channel BN sum / sum-of-squares partials (wave-reduced, then atomics).
// One wave per block; grid-strides over 16-row tiles.
// ---------------------------------------------------------------------------
__global__ void node_mlp_kernel(const float* __restrict__ h,
                                const float* __restrict__ aggr,
                                const float* __restrict__ pkw1,
                                const float* __restrict__ b1,
                                const float* __restrict__ pkw2,
                                const float* __restrict__ b2,
                                float* __restrict__ z,
                                float* __restrict__ bnstat,
                                int n, int ntiles) {
  __shared__ float Z[16 * HDIM];
  __shared__ float T[16 * HDIM];
  const int lane = threadIdx.x & 31;
  const int m    = lane & 15;
  const int half = lane >> 4;
  float bnS[8], bnQ[8];
#pragma unroll
  for (int i = 0; i < 8; ++i) { bnS[i] = 0.f; bnQ[i] = 0.f; }

  for (int tile = blockIdx.x; tile < ntiles; tile += gridDim.x) {
    const int r0 = tile * 16;
    const bool fullN = (r0 + 16 <= n);       // wave-uniform
    // stage zin = h + aggr (row-major in LDS); clamp reads, pad rows unused.
    const long long base = (long long)r0 * HDIM;
    const long long maxi = (long long)n * HDIM - 2;
    for (int i = lane * 2; i < 16 * HDIM; i += 64) {
      const long long gi = (fullN || base + i <= maxi) ? base + i : maxi;
      v2f hv = *(const v2f*)&h[gi];
      v2f av = *(const v2f*)&aggr[gi];
      Z[i]     = hv.x + av.x;
      Z[i + 1] = hv.y + av.y;
    }
    __syncthreads();
    // t1 = relu(Z @ w1 + b1)
#pragma unroll
    for (int nt = 0; nt < 8; ++nt) {
      v8f acc = {};
      const int nn = nt * 16 + m;
      for (int ks = 0; ks < 32; ++ks) {
        const int k = ks * 4 + half * 2;
        v2f a = *(const v2f*)&Z[m * HDIM + k];
        acc = wmma_f32_k4(a, bfrag(pkw1, 32, nt, ks, lane), acc);
      }
      const float bias = b1[nn];
#pragma unroll
      for (int v = 0; v < 8; ++v) {
        float t = acc[v] + bias;
        T[(v + half * 8) * HDIM + nn] = t > 0.f ? t : 0.f;
      }
    }
    __syncthreads();
    // z = T @ w2 + b2, plus BN partials
#pragma unroll
    for (int nt = 0; nt < 8; ++nt) {
      v8f acc = {};
      const int nn = nt * 16 + m;
      for (int ks = 0; ks < 32; ++ks) {
        const int k = ks * 4 + half * 2;
        v2f a = *(const v2f*)&T[m * HDIM + k];
        acc = wmma_f32_k4(a, bfrag(pkw2, 32, nt, ks, lane), acc);
      }
      const float bias = b2[nn];
      if (fullN) {
#pragma unroll
        for (int v = 0; v < 8; ++v) {
          const float zv = acc[v] + bias;
          z[(r0 + v + half * 8) * HDIM + nn] = zv;
          bnS[nt] += zv;
          bnQ[nt] += zv * zv;
        }
      } else {
#pragma unroll
        for (int v = 0; v < 8; ++v) {
          const int row = r0 + v + half * 8;
          if (row < n) {
            const float zv = acc[v] + bias;
            z[row * HDIM + nn] = zv;
            bnS[nt] += zv;
            bnQ[nt] += zv * zv;
          }
        }
      }
    }
    __syncthreads();
  }
  // lanes t and t^16 hold the same channel (nt*16 + (t&15)): combine + atomic.
#pragma unroll
  for (int nt = 0; nt < 8; ++nt) {
    float s = bnS[nt] + __shfl_xor(bnS[nt], 16, 32);
    float q = bnQ[nt] + __shfl_xor(bnQ[nt], 16, 32);
    if (half == 0) {
      atomicAdd(&bnstat[nt * 16 + m], s);
      atomicAdd(&bnstat[HDIM + nt * 16 + m], q);
    }
  }
}

// ---------------------------------------------------------------------------
// h = relu(batchnorm(z)) with batch statistics from bnstat (sum, sumsq).
// ---------------------------------------------------------------------------
__global__ void bn_relu_kernel(const float* __restrict__ z,
                               const float* __restrict__ bnstat,
                               const float* __restrict__ gamma,
                               const float* __restrict__ beta,
                               float* __restrict__ h, int n) {
  const float invN = 1.0f / (float)n;
  const long long total  = (long long)n * HDIM;
  const long long stride = (long long)gridDim.x * blockDim.x;
  for (long long idx = (long long)blockIdx.x * blockDim.x + threadIdx.x;
       idx < total; idx += stride) {
    const int c = (int)(idx & (HDIM - 1));
    const float mean = bnstat[c] * invN;
    const float var  = fmaxf(bnstat[HDIM + c] * invN - mean * mean, 0.f);
    float v = (z[idx] - mean) * rsqrtf(var + 1e-5f) * gamma[c] + beta[c];
    h[idx] = v > 0.f ? v : 0.f;
  }
}

// ---------------------------------------------------------------------------
// Global mean-pool accumulation: gsum[batch[i]] += h[i]; gcnt[batch[i]] += 1.
// ---------------------------------------------------------------------------
__global__ void pool_kernel(const float* __restrict__ h,
                            const int* __restrict__ batch,
                            float* __restrict__ gsum,
                            float* __restrict__ gcnt, int n) {
  const long long total  = (long long)n * HDIM;
  const long long stride = (long long)gridDim.x * blockDim.x;
  for (long long idx = (long long)blockIdx.x * blockDim.x + threadIdx.x;
       idx < total; idx += stride) {
    const int node = (int)(idx >> 7);
    const int c    = (int)(idx & (HDIM - 1));
    atomicAdd(&gsum[batch[node] * HDIM + c], h[idx]);
  }
  for (int i = blockIdx.x * blockDim.x + threadIdx.x; i < n;
       i += gridDim.x * blockDim.x)
    atomicAdd(&gcnt[batch[i]], 1.0f);
}

// ---------------------------------------------------------------------------
// ext = relu(externals @ w1 + b1) @ w2 + b2.  One block (128 thr) per graph.
// ---------------------------------------------------------------------------
__global__ void ext_mlp_kernel(const float* __restrict__ ext,
                               const float* __restrict__ w1,
                               const float* __restrict__ b1,
                               const float* __restrict__ w2,
                               const float* __restrict__ b2,
                               float* __restrict__ out) {
  __shared__ float hid[HDIM];
  const int g = blockIdx.x;
  const int j = threadIdx.x;
  float acc = b1[j];
#pragma unroll
  for (int k = 0; k < 8; ++k) acc += ext[g * 8 + k] * w1[k * HDIM + j];
  hid[j] = acc > 0.f ? acc : 0.f;
  __syncthreads();
  float o = b2[j];
  for (int k = 0; k < HDIM; ++k) o += hid[k] * w2[k * HDIM + j];
  out[g * HDIM + j] = o;
}

// ---------------------------------------------------------------------------
// Final head: comb = [gsum/cnt, ext]; out = relu(comb@w1+b1) @ w2 + b2.
// ---------------------------------------------------------------------------
__global__ void regressor_kernel(const float* __restrict__ gsum,
                                 const float* __restrict__ gcnt,
                                 const float* __restrict__ extf,
                                 const float* __restrict__ w1,
                                 const float* __restrict__ b1,
                                 const float* __restrict__ w2,
                                 const float* __restrict__ b2,
                                 float* __restrict__ out) {
  __shared__ float comb[2 * HDIM];
  __shared__ float red[HDIM];
  const int g = blockIdx.x;
  const int j = threadIdx.x;
  const float cnt = fmaxf(gcnt[g], 1.0f);
  comb[j]        = gsum[g * HDIM + j] / cnt;
  comb[HDIM + j] = extf[g * HDIM + j];
  __syncthreads();
  float acc = b1[j];
  for (int k = 0; k < 2 * HDIM; ++k) acc += comb[k] * w1[k * HDIM + j];
  acc = acc > 0.f ? acc : 0.f;
  red[j] = acc * w2[j];
  __syncthreads();
  for (int s = HDIM / 2; s > 0; s >>= 1) {
    if (j < s) red[j] += red[j + s];
    __syncthreads();
  }
  if (j == 0) out[g] = red[0] + b2[0];
}

// ---------------------------------------------------------------------------
extern "C" void kernel_launch(void* const* d_in, const int* in_sizes, int n_in,
                              void* d_out, int out_size, void* d_ws,
                              size_t ws_size, hipStream_t stream) {
  const float* x         = (const float*)d_in[0];
  const float* edge_attr = (const float*)d_in[1];
  const float* externals = (const float*)d_in[2];
  const float* node_w    = (const float*)d_in[3];
  const float* node_b    = (const float*)d_in[4];
  const float* ee_w1     = (const float*)d_in[5];
  const float* ee_b1     = (const float*)d_in[6];
  const float* ee_w2     = (const float*)d_in[7];
  const float* ee_b2     = (const float*)d_in[8];
  const float* conv_w1   = (const float*)d_in[9];
  const float* conv_b1   = (const float*)d_in[10];
  const float* conv_w2   = (const float*)d_in[11];
  const float* conv_b2   = (const float*)d_in[12];
  const float* bn_gamma  = (const float*)d_in[13];
  const float* bn_beta   = (const float*)d_in[14];
  const float* ext_w1    = (const float*)d_in[15];
  const float* ext_b1    = (const float*)d_in[16];
  const float* ext_w2    = (const float*)d_in[17];
  const float* ext_b2    = (const float*)d_in[18];
  const float* reg_w1    = (const float*)d_in[19];
  const float* reg_b1    = (const float*)d_in[20];
  const float* reg_w2    = (const float*)d_in[21];
  const float* reg_b2    = (const float*)d_in[22];
  const int* edge_index  = (const int*)d_in[23];
  const int* batch       = (const int*)d_in[24];

  const int N = in_sizes[0] / 32;            // node feature dim 32
  const int E = in_sizes[1] / 16;            // edge feature dim 16
  const int G = in_sizes[2] / 8;             // external feature dim 8
  const int L = in_sizes[9] / (HDIM * HDIM); // stacked conv layers

  // workspace: h | aggr | z | bnstat | gsum | gcnt | extf | packed weights
  float* ws      = (float*)d_ws;
  float* h       = ws;
  float* aggr    = h + (size_t)N * HDIM;
  float* z       = aggr + (size_t)N * HDIM;
  float* bnstat  = z + (size_t)N * HDIM;
  float* gsum    = bnstat + 2 * HDIM;
  float* gcnt    = gsum + (size_t)G * HDIM;
  float* extf    = gcnt + G;
  float* pk_node = extf + (size_t)G * HDIM;       // 32*128
  float* pk_ew1  = pk_node + 32 * HDIM;           // 16*128
  float* pk_ew2  = pk_ew1 + 16 * HDIM;            // 128*128
  float* pk_cw1  = pk_ew2 + HDIM * HDIM;          // L*128*128
  float* pk_cw2  = pk_cw1 + (size_t)L * HDIM * HDIM;

  const int* src = edge_index;
  const int* dst = edge_index + E;

  const int tilesN = (N + 15) / 16;
  const int tilesE = (E + 15) / 16;

  // one-time (per launch) weight repacking into B-fragment order
  pack_b_kernel<<<16, 256, 0, stream>>>(node_w, pk_node, 32);
  pack_b_kernel<<<8, 256, 0, stream>>>(ee_w1, pk_ew1, 16);
  pack_b_kernel<<<32, 256, 0, stream>>>(ee_w2, pk_ew2, 128);
  for (int l = 0; l < L; ++l) {
    pack_b_kernel<<<32, 256, 0, stream>>>(conv_w1 + (size_t)l * HDIM * HDIM,
                                          pk_cw1 + (size_t)l * HDIM * HDIM,
                                          128);
    pack_b_kernel<<<32, 256, 0, stream>>>(conv_w2 + (size_t)l * HDIM * HDIM,
                                          pk_cw2 + (size_t)l * HDIM * HDIM,
                                          128);
  }

  node_encode_kernel<<<(tilesN + 3) / 4, 128, 0, stream>>>(x, pk_node, node_b,
                                                           h, N);

  for (int l = 0; l < L; ++l) {
    hipMemsetAsync(aggr, 0, (size_t)N * HDIM * sizeof(float), stream);
    hipMemsetAsync(bnstat, 0, 2 * HDIM * sizeof(float), stream);
    msg_aggr_kernel<<<(tilesE + 3) / 4, 128, 0, stream>>>(
        edge_attr, pk_ew1, ee_b1, pk_ew2, ee_b2, h, src, dst, aggr, E, tilesE);
    int nb = tilesN < 625 ? tilesN : 625;
    node_mlp_kernel<<<nb, 32, 0, stream>>>(
        h, aggr, pk_cw1 + (size_t)l * HDIM * HDIM, conv_b1 + l * HDIM,
        pk_cw2 + (size_t)l * HDIM * HDIM, conv_b2 + l * HDIM, z, bnstat, N,
        tilesN);
    bn_relu_kernel<<<1024, 256, 0, stream>>>(z, bnstat, bn_gamma + l * HDIM,
                                             bn_beta + l * HDIM, h, N);
  }

  hipMemsetAsync(gsum, 0, ((size_t)G * HDIM + G) * sizeof(float), stream);
  pool_kernel<<<1024, 256, 0, stream>>>(h, batch, gsum, gcnt, N);
  ext_mlp_kernel<<<G, HDIM, 0, stream>>>(externals, ext_w1, ext_b1, ext_w2,
                                         ext_b2, extf);
  regressor_kernel<<<G, HDIM, 0, stream>>>(gsum, gcnt, extf, reg_w1, reg_b1,
                                           reg_w2, reg_b2, (float*)d_out);
}